// CPCLossV2_10557029613703
// MI455X (gfx1250) — compile-verified
//
#include <hip/hip_runtime.h>
#include <hip/hip_bf16.h>
#include <math.h>

// Problem constants (from the reference's setup_inputs)
#define NG     4096   // n groups
#define KPOS   4
#define HDIM   256
#define MNEG   64
#define KD     ((KPOS - 1) * HDIM)   // 768 = GEMM K
#define NLOGIT (MNEG + 1)            // 65

typedef __attribute__((ext_vector_type(2))) float v2f;
typedef __attribute__((ext_vector_type(8))) float v8f;

// ---------------------------------------------------------------------------
// Kernel 1: predicts = hist_x @ W + b   (M=4096, K=768, N=256), f32 WMMA.
// hist_x[g] is the contiguous slice embeddings[g*1024 : g*1024 + 768].
// One wave32 computes one 16x16 tile of C via V_WMMA_F32_16X16X4_F32.
//
// A (16x4) lane layout (ISA 7.12.2, 32-bit A 16x4):
//   lanes 0-15:  V0 = A[lane, K=0], V1 = A[lane, K=1]
//   lanes 16-31: V0 = A[lane-16, K=2], V1 = A[lane-16, K=3]
// B (4x16): rows striped across lanes within a VGPR (mirror of A):
//   lanes 0-15:  V0 = B[K=0, lane], V1 = B[K=1, lane]
//   lanes 16-31: V0 = B[K=2, lane-16], V1 = B[K=3, lane-16]
// C/D (16x16, 8 VGPRs): VGPR r holds M = r + 8*(lane>=16), N = lane & 15.
// ---------------------------------------------------------------------------
__global__ __launch_bounds__(128) void cpc_predicts_gemm(
    const float* __restrict__ emb,   // (NG*KPOS, HDIM)
    const float* __restrict__ W,     // (KD, HDIM)
    const float* __restrict__ bias,  // (HDIM,)
    float* __restrict__ predicts)    // (NG, HDIM)  -> d_ws
{
  const int wave  = threadIdx.x >> 5;
  const int lane  = threadIdx.x & 31;
  const int tile  = blockIdx.x * 4 + wave;      // 0 .. 4095
  const int mt    = tile >> 4;                  // 0 .. 255  (group tile)
  const int nt    = tile & 15;                  // 0 .. 15   (col tile)
  const int m0    = mt * 16;
  const int n0    = nt * 16;

  const int lh    = lane & 15;                  // 0..15
  const int khalf = lane >> 4;                  // 0 or 1 -> k offset 2*khalf

  // hist_x row (m0+lh): contiguous 768 floats, row stride 1024 floats
  const float* __restrict__ arow = emb + (size_t)(m0 + lh) * (KPOS * HDIM);
  const float* __restrict__ bcol = W + (size_t)(n0 + lh);

  v8f acc = {0.f, 0.f, 0.f, 0.f, 0.f, 0.f, 0.f, 0.f};

#pragma unroll 4
  for (int k0 = 0; k0 < KD; k0 += 4) {
    const int ka = k0 + 2 * khalf;
    v2f a;                                      // contiguous pair -> b64 load
    a.x = arow[ka + 0];
    a.y = arow[ka + 1];
    v2f b;
    b.x = bcol[(size_t)(ka + 0) * HDIM];
    b.y = bcol[(size_t)(ka + 1) * HDIM];
    acc = __builtin_amdgcn_wmma_f32_16x16x4_f32(
        /*neg_a=*/false, a, /*neg_b=*/false, b,
        /*c_mod=*/(short)0, acc, /*reuse_a=*/false, /*reuse_b=*/false);
  }

  const float bv    = bias[n0 + lh];
  const int   rbase = m0 + (khalf ? 8 : 0);
#pragma unroll
  for (int r = 0; r < 8; ++r)
    predicts[(size_t)(rbase + r) * HDIM + (n0 + lh)] = acc[r] + bv;
}

// ---------------------------------------------------------------------------
// Kernel 2: per-group logits + log-softmax loss contribution.
//   logit[0]   = dot(predicts[g], embeddings[4g+3])            (positive)
//   logit[1+j] = dot(predicts[g], embeddings[fix(perm[g,j])])  (negatives)
//   fix(p) = p + (p >= 4g ? 4 : 0)   -- closed form of the cand/where gather
//   loss += (logsumexp(logits) - logit[0]) / NG
// One block (256 threads = 8 wave32) per group. Gathers hit L2 (table=16.8MB).
// ---------------------------------------------------------------------------
__global__ __launch_bounds__(256) void cpc_logits_loss(
    const float* __restrict__ emb,
    const float* __restrict__ predicts,
    const int*   __restrict__ perm,   // (NG, MNEG)
    float* __restrict__ out)          // scalar accumulator (pre-zeroed)
{
  const int g    = blockIdx.x;
  const int tid  = threadIdx.x;
  const int wave = tid >> 5;
  const int lane = tid & 31;

  __shared__ float pred_s[HDIM];
  __shared__ float logits_s[NLOGIT];

  pred_s[tid] = predicts[(size_t)g * HDIM + tid];
  __syncthreads();

  for (int s = wave; s < NLOGIT; s += 8) {
    const float* __restrict__ row;
    if (s == 0) {
      row = emb + ((size_t)g * KPOS + (KPOS - 1)) * HDIM;      // hist_y
    } else {
      int p   = perm[(size_t)g * MNEG + (s - 1)];
      int idx = p + ((p >= KPOS * g) ? KPOS : 0);
      row = emb + (size_t)idx * HDIM;
    }
    float a = 0.f;
#pragma unroll
    for (int e = lane; e < HDIM; e += 32)
      a = fmaf(row[e], pred_s[e], a);
    // wave32 butterfly reduction
#pragma unroll
    for (int msk = 16; msk >= 1; msk >>= 1)
      a += __shfl_xor(a, msk, 32);
    if (lane == 0) logits_s[s] = a;
  }
  __syncthreads();

  if (wave == 0) {
    float mx = -INFINITY;
    for (int s = lane; s < NLOGIT; s += 32)
      mx = fmaxf(mx, logits_s[s]);
#pragma unroll
    for (int msk = 16; msk >= 1; msk >>= 1)
      mx = fmaxf(mx, __shfl_xor(mx, msk, 32));

    float sum = 0.f;
    for (int s = lane; s < NLOGIT; s += 32)
      sum += __expf(logits_s[s] - mx);
#pragma unroll
    for (int msk = 16; msk >= 1; msk >>= 1)
      sum += __shfl_xor(sum, msk, 32);

    if (lane == 0) {
      const float lse = mx + __logf(sum);
      atomicAdd(out, (lse - logits_s[0]) * (1.0f / (float)NG));
    }
  }
}

// ---------------------------------------------------------------------------
// Launcher. Inputs (setup_inputs order):
//   0: embeddings f32 (16384,256)  1: W f32 (768,256)  2: b f32 (256,)
//   3: target i32 (16384,)         4: perm i32 (4096,64)
//   5: k_pos_samples i32 scalar    6: m_neg_samples i32 scalar
// Output: 1 x f32 scalar loss. d_ws: predicts (4096*256 f32 = 4 MB).
// ---------------------------------------------------------------------------
extern "C" void kernel_launch(void* const* d_in, const int* in_sizes, int n_in,
                              void* d_out, int out_size, void* d_ws, size_t ws_size,
                              hipStream_t stream) {
  (void)in_sizes; (void)n_in; (void)out_size; (void)ws_size;
  const float* emb  = (const float*)d_in[0];
  const float* W    = (const float*)d_in[1];
  const float* bias = (const float*)d_in[2];
  const int*   perm = (const int*)d_in[4];

  float* predicts = (float*)d_ws;
  float* out      = (float*)d_out;

  // Zero the scalar loss accumulator (memset node is graph-capturable).
  hipMemsetAsync(out, 0, sizeof(float), stream);

  // GEMM: 4096 wave-tiles, 4 waves (128 threads) per block.
  cpc_predicts_gemm<<<NG * (HDIM / 16) / 4 / 16 * 16, 128, 0, stream>>>(
      emb, W, bias, predicts);

  // Logits + loss: one block per group.
  cpc_logits_loss<<<NG, 256, 0, stream>>>(emb, predicts, perm, out);
}